// GCN_46626164965966
// MI455X (gfx1250) — compile-verified
//
#include <hip/hip_runtime.h>

typedef __attribute__((ext_vector_type(2))) float v2f;
typedef __attribute__((ext_vector_type(8))) float v8f;

#define HID 128

// ---------------- utility kernels ----------------

__global__ __launch_bounds__(256) void k_zero(float* p, int n) {
  int i = blockIdx.x * blockDim.x + threadIdx.x;
  if (i < n) p[i] = 0.0f;
}

__global__ __launch_bounds__(256) void k_deg_accum(const int* __restrict__ dst,
                                                   float* __restrict__ deg, int E) {
  int e = blockIdx.x * blockDim.x + threadIdx.x;
  if (e < E) atomicAdd(&deg[dst[e]], 1.0f);
}

__global__ __launch_bounds__(256) void k_deg_to_dinv(float* d, int n) {
  int i = blockIdx.x * blockDim.x + threadIdx.x;
  if (i < n) d[i] = rsqrtf(d[i] + 1.0f);  // self-loop included
}

// ---------------- weight repack into WMMA-fragment order ----------------
// Wp[p*128 + n] = (W[k][n], W[k+1][n]) with k = (p>>1)*4 + (p&1)*2, zero-padded.
// B fragment load in the GEMM becomes a single global_load_b64.

__global__ __launch_bounds__(256) void k_pack_w(const float* __restrict__ W,
                                                float2* __restrict__ Wp,
                                                int K, int Kpad) {
  int idx = blockIdx.x * blockDim.x + threadIdx.x;
  int total = (Kpad >> 1) * HID;
  if (idx >= total) return;
  int p = idx >> 7;
  int n = idx & 127;
  int k = (p >> 1) * 4 + (p & 1) * 2;
  float a = (k < K) ? W[k * HID + n] : 0.0f;
  float b = (k + 1 < K) ? W[(k + 1) * HID + n] : 0.0f;
  Wp[idx] = make_float2(a, b);
}

// ---------------- fused fp32 WMMA GEMM + GCN epilogue ----------------
// h[Mx128] = act(A)[MxK] @ W[Kx128]            (act = ReLU if RELU, else id)
// agg[Mx128] = h * dinv[row]^2 + bias[col]     (self-loop + bias init)
// Block = 256 threads = 8 waves; each wave: one 16-col N-tile, MT row-subtiles.
// 50000 = 625 * 80 (MT = 5), exact. K compile-time -> branch-free inner loop.
template <int K, int MT, bool RELU>
__global__ __launch_bounds__(256) void k_gemm_fused(
    const float* __restrict__ A, const float2* __restrict__ Wp,
    const float* __restrict__ bias, const float* __restrict__ dinv,
    float* __restrict__ h, float* __restrict__ agg) {
  constexpr int Kpad = (K + 3) & ~3;
  constexpr int LDA = Kpad + 4;  // 8B-aligned even-k pairs, bank-conflict-free
  __shared__ float lA[16 * MT][LDA];
  const int m0 = blockIdx.x * (16 * MT);

  // cooperative A-tile load with optional fused ReLU, zero-pad K -> Kpad
  for (int idx = threadIdx.x; idx < 16 * MT * Kpad; idx += 256) {
    int r = idx / Kpad;
    int k = idx - r * Kpad;
    float v = (k < K) ? A[(size_t)(m0 + r) * K + k] : 0.0f;
    if (RELU) v = fmaxf(v, 0.0f);
    lA[r][k] = v;
  }
  __syncthreads();

  const int lane = threadIdx.x & 31;
  const int wid  = threadIdx.x >> 5;
  const int n0   = wid * 16;   // this wave's N-tile
  const int kh   = lane >> 4;  // half-wave: K pair (0,1) vs (2,3)
  const int l15  = lane & 15;  // M index for A frag, N index for B frag

  v8f c[MT];
#pragma unroll
  for (int t = 0; t < MT; ++t) c[t] = (v8f){};

#pragma unroll 4
  for (int k0 = 0; k0 < Kpad; k0 += 4) {
    // packed B fragment: one b64 load
    v2f b = *(const v2f*)&Wp[(size_t)((k0 >> 1) + kh) * HID + n0 + l15];
#pragma unroll
    for (int t = 0; t < MT; ++t) {
      v2f a;
      a.x = lA[t * 16 + l15][k0 + 2 * kh];  // consecutive pair -> ds_load_b64
      a.y = lA[t * 16 + l15][k0 + 2 * kh + 1];
      c[t] = __builtin_amdgcn_wmma_f32_16x16x4_f32(false, a, false, b, (short)0,
                                                   c[t], false, false);
    }
  }

  // Epilogue: C/D layout VGPR i -> row i (lanes 0-15) / i+8 (lanes 16-31).
  const float bn = bias[n0 + l15];
#pragma unroll
  for (int t = 0; t < MT; ++t) {
#pragma unroll
    for (int i = 0; i < 8; ++i) {
      int row = m0 + t * 16 + i + 8 * kh;
      float dv = dinv[row];  // half-wave-uniform broadcast
      size_t o = (size_t)row * HID + n0 + l15;
      float hv = c[t][i];
      h[o] = hv;
      agg[o] = fmaf(hv, dv * dv, bn);
    }
  }
}

// ---------------- edge scatter: one wave32 per edge ----------------

__global__ __launch_bounds__(256) void k_edge_scatter(const int* __restrict__ src,
                                                      const int* __restrict__ dst,
                                                      const float* __restrict__ h,
                                                      const float* __restrict__ dinv,
                                                      float* __restrict__ agg, int E) {
  int gid = blockIdx.x * blockDim.x + threadIdx.x;
  int e = gid >> 5;
  int lane = gid & 31;
  if (e >= E) return;
  // prefetch the gather row of an edge well ahead (hides L2 latency)
  int e2 = e + 64;
  if (e2 < E) {
    int s2 = src[e2];
    __builtin_prefetch(h + (size_t)s2 * HID + lane * 4, 0, 3);
  }
  int s = src[e], d = dst[e];  // wave-uniform -> scalarized
  float nrm = dinv[s] * dinv[d];
  const float4* hs = (const float4*)(h + (size_t)s * HID);
  float4 v = hs[lane];  // 32 lanes * 4 floats = 128 features
  float* ad = agg + (size_t)d * HID + lane * 4;
  atomicAdd(ad + 0, v.x * nrm);
  atomicAdd(ad + 1, v.y * nrm);
  atomicAdd(ad + 2, v.z * nrm);
  atomicAdd(ad + 3, v.w * nrm);
}

// ---------------- pooling + final linear ----------------

__global__ __launch_bounds__(256) void k_pool_accum(const float* __restrict__ h,
                                                    const int* __restrict__ batch,
                                                    float* __restrict__ sums, int N) {
  int gid = blockIdx.x * blockDim.x + threadIdx.x;
  int node = gid >> 5;
  int c = gid & 31;
  if (node >= N) return;
  int g = batch[node];
  const float4* h4 = (const float4*)(h + (size_t)node * HID);
  float4 v = h4[c];
  float* sp = sums + (size_t)g * HID + c * 4;
  atomicAdd(sp + 0, v.x);
  atomicAdd(sp + 1, v.y);
  atomicAdd(sp + 2, v.z);
  atomicAdd(sp + 3, v.w);
}

__global__ __launch_bounds__(256) void k_pool_cnt(const int* __restrict__ batch,
                                                  float* __restrict__ cnt, int N) {
  int i = blockIdx.x * blockDim.x + threadIdx.x;
  if (i < N) atomicAdd(&cnt[batch[i]], 1.0f);
}

__global__ __launch_bounds__(256) void k_final_lin(const float* __restrict__ sums,
                                                   const float* __restrict__ cnt,
                                                   const float* __restrict__ linW,
                                                   const float* __restrict__ linb,
                                                   float* __restrict__ out, int G, int C) {
  int idx = blockIdx.x * blockDim.x + threadIdx.x;
  if (idx >= G * C) return;
  int g = idx / C;
  int c = idx - g * C;
  float inv = 1.0f / fmaxf(cnt[g], 1.0f);
  float acc = linb[c];
  for (int f = 0; f < HID; ++f)
    acc = fmaf(sums[(size_t)g * HID + f] * inv, linW[f * C + c], acc);
  out[idx] = acc;
}

// ---------------- launch ----------------

extern "C" void kernel_launch(void* const* d_in, const int* in_sizes, int n_in,
                              void* d_out, int out_size, void* d_ws, size_t ws_size,
                              hipStream_t stream) {
  (void)n_in; (void)out_size; (void)ws_size;
  const float* x    = (const float*)d_in[0];
  const float* W1   = (const float*)d_in[1];
  const float* b1   = (const float*)d_in[2];
  const float* W2   = (const float*)d_in[3];
  const float* b2   = (const float*)d_in[4];
  const float* W3   = (const float*)d_in[5];
  const float* b3   = (const float*)d_in[6];
  const float* W4   = (const float*)d_in[7];
  const float* b4   = (const float*)d_in[8];
  const float* linW = (const float*)d_in[9];
  const float* linb = (const float*)d_in[10];
  const int* edge   = (const int*)d_in[11];
  const int* batch  = (const int*)d_in[12];

  const int N = in_sizes[0] / 3;   // 50000 nodes (NUM_FEAT = 3)
  const int E = in_sizes[11] / 2;  // 600000 edges
  const int G = 512;               // NUM_GRAPHS (fixed in reference)
  const int C = 2;                 // NUM_CLASSES

  const int* src = edge;
  const int* dst = edge + E;

  float* ws     = (float*)d_ws;
  float* d_h    = ws;                        // N*128
  float* d_agg  = d_h + (size_t)N * HID;     // N*128
  float* d_dinv = d_agg + (size_t)N * HID;   // N (deg, then rsqrt in place)
  float* d_sum  = d_dinv + N;                // G*128
  float* d_cnt  = d_sum + (size_t)G * HID;   // G (contiguous with sums)
  float* wp     = d_cnt + G;                 // packed weights
  float2* d_wp0 = (float2*)wp;               // Kpad=4  : 2*128 float2
  float2* d_wp1 = d_wp0 + 2 * HID;           // Kpad=128: 64*128 float2 each
  float2* d_wp2 = d_wp1 + 64 * HID;
  float2* d_wp3 = d_wp2 + 64 * HID;

  const int NT = 256;
  auto blks = [](long long n, int t) { return (int)((n + t - 1) / t); };

  // degrees + dinv (shared by all layers)
  k_zero<<<blks(N, NT), NT, 0, stream>>>(d_dinv, N);
  k_deg_accum<<<blks(E, NT), NT, 0, stream>>>(dst, d_dinv, E);
  k_deg_to_dinv<<<blks(N, NT), NT, 0, stream>>>(d_dinv, N);

  // pack weights into fragment order (one-time, tiny)
  k_pack_w<<<blks(2 * HID, NT), NT, 0, stream>>>(W1, d_wp0, 3, 4);
  k_pack_w<<<blks(64 * HID, NT), NT, 0, stream>>>(W2, d_wp1, HID, HID);
  k_pack_w<<<blks(64 * HID, NT), NT, 0, stream>>>(W3, d_wp2, HID, HID);
  k_pack_w<<<blks(64 * HID, NT), NT, 0, stream>>>(W4, d_wp3, HID, HID);

  const float* bs[4] = {b1, b2, b3, b4};
  float2* wps[4] = {d_wp0, d_wp1, d_wp2, d_wp3};
  const int gemmGrid = N / 80;  // MT = 5 -> 80 rows per block (625 blocks)

  const float* act = x;
  for (int l = 0; l < 4; ++l) {
    if (l == 0)
      k_gemm_fused<3, 5, false><<<gemmGrid, NT, 0, stream>>>(act, wps[l], bs[l],
                                                             d_dinv, d_h, d_agg);
    else
      k_gemm_fused<HID, 5, true><<<gemmGrid, NT, 0, stream>>>(act, wps[l], bs[l],
                                                              d_dinv, d_h, d_agg);
    k_edge_scatter<<<blks((long long)E * 32, NT), NT, 0, stream>>>(src, dst, d_h,
                                                                   d_dinv, d_agg, E);
    act = d_agg;
  }

  // global mean pool + final linear (layer-4 agg has no ReLU, as in reference)
  k_zero<<<blks((long long)G * HID + G, NT), NT, 0, stream>>>(d_sum, G * HID + G);
  k_pool_accum<<<blks((long long)N * 32, NT), NT, 0, stream>>>(d_agg, batch, d_sum, N);
  k_pool_cnt<<<blks(N, NT), NT, 0, stream>>>(batch, d_cnt, N);
  k_final_lin<<<blks(G * C, NT), NT, 0, stream>>>(d_sum, d_cnt, linW, linb,
                                                  (float*)d_out, G, C);
}